// FPSSampler_22943715295397
// MI455X (gfx1250) — compile-verified
//
#include <hip/hip_runtime.h>
#include <stdint.h>

// ---------------------------------------------------------------------------
// Farthest-point sampling for MI455X (gfx1250).
//
// B=32, N=131072, k=2048.  Sequentially dependent over k, so the design goal
// is minimum per-iteration latency:
//   * all per-point state (xyz + running min-dist) lives in VGPRs
//   * each batch is split across SPLIT=8 workgroups (256 WGs -> fills the
//     chip; 8x less per-WGP work per iteration than 1 WG/batch)
//   * cross-WG argmax via parity-double-buffered L2 slots + device-scope
//     release/acquire atomic arrive counter (lane-parallel slot combine)
//   * one-time point staging uses the CDNA5 async global->LDS path
//     (global_load_async_to_lds_b128 + s_wait_asynccnt)
// WMMA is not applicable: there is no batched GEMM shape in FPS (each step
// depends on the previous argmax).
// ---------------------------------------------------------------------------

#define B_    32
#define N_    131072
#define K_    2048
#define SPLIT 8
#define NLOC  (N_ / SPLIT)        // 16384 points per workgroup
#define TPB   1024
#define PPT   (NLOC / TPB)        // 16 points per thread (register resident)
#define RAW_FLOATS (NLOC * 3)     // 49152 floats = 196608 B staged in LDS
#define LDS_BYTES  (RAW_FLOATS * 4 + 512)

typedef unsigned long long u64;
typedef unsigned int u32;

__global__ void fps_init(u32* cnt) {
  if (threadIdx.x < B_) cnt[threadIdx.x] = 0u;
}

__device__ __forceinline__ u64 pack2f(float a, float b) {
  return ((u64)__float_as_uint(b) << 32) | (u64)__float_as_uint(a);
}

__global__ __launch_bounds__(TPB) void fps_kernel(const float* __restrict__ x,
                                                  float* __restrict__ out,
                                                  u32* __restrict__ cnt,
                                                  u64* __restrict__ slots) {
  const int b     = blockIdx.x / SPLIT;
  const int split = blockIdx.x % SPLIT;
  const int tid   = threadIdx.x;
  const int base  = split * NLOC;          // this WG's first point (within batch)

  extern __shared__ unsigned char smem[];
  float* sRaw = (float*)smem;                              // staged points
  float* sRv  = (float*)(smem + RAW_FLOATS * 4);           // per-wave max val
  int*   sRi  = (int*)  (smem + RAW_FLOATS * 4 + 128);     // per-wave max idx
  float* sC   = (float*)(smem + RAW_FLOATS * 4 + 256);     // broadcast centroid

  const float* gsrc = x + ((size_t)b * N_ + (size_t)base) * 3;

  // ---- one-time async stage: 192KB global -> LDS (ASYNCcnt path) ----------
  {
    u32 ldsBase = (u32)(uintptr_t)sRaw;    // low 32 bits of LDS-aperture addr
    const int chunks = RAW_FLOATS / 4;     // 12288 x b128, 16B-aligned
    for (int i = tid; i < chunks; i += TPB) {
      u32 loff = ldsBase + (u32)i * 16u;
      u64 ga   = (u64)(uintptr_t)(gsrc + (size_t)i * 4);
      asm volatile("global_load_async_to_lds_b128 %0, %1, off"
                   :: "v"(loff), "v"(ga) : "memory");
    }
    asm volatile("s_wait_asynccnt 0x0" ::: "memory");
  }
  // First centroid is point 0 of the batch (reference starts at index 0).
  if (tid == 0) {
    sC[0] = x[(size_t)b * N_ * 3 + 0];
    sC[1] = x[(size_t)b * N_ * 3 + 1];
    sC[2] = x[(size_t)b * N_ * 3 + 2];
  }
  __syncthreads();

  // ---- lift this thread's 16 points into registers (conflict-free: bank =
  //      3*lane mod 64, 3 is invertible mod 64) ------------------------------
  float px[PPT], py[PPT], pz[PPT], dd[PPT];
#pragma unroll
  for (int q = 0; q < PPT; ++q) {
    int l = q * TPB + tid;
    px[q] = sRaw[l * 3 + 0];
    py[q] = sRaw[l * 3 + 1];
    pz[q] = sRaw[l * 3 + 2];
    dd[q] = __builtin_inff();
  }

  for (int j = 0; j < K_; ++j) {
    const float cx = sC[0], cy = sC[1], cz = sC[2];
    if (split == 0 && tid == 0) {          // emit selected point of this step
      float* o = out + ((size_t)b * K_ + j) * 3;
      o[0] = cx; o[1] = cy; o[2] = cz;
    }

    // ---- register-resident min-update + local argmax ----------------------
    float bv = -1.0f; int bi = 0;
#pragma unroll
    for (int q = 0; q < PPT; ++q) {
      float dx = px[q] - cx, dy = py[q] - cy, dz = pz[q] - cz;
      float d  = __fmaf_rn(dx, dx, __fmaf_rn(dy, dy, dz * dz));
      float nd = fminf(dd[q], d);
      dd[q] = nd;
      // idx grows with q, so strict '>' keeps the first (lowest-index) max.
      if (nd > bv) { bv = nd; bi = base + q * TPB + tid; }
    }

    // ---- wave32 shuffle reduce (tie-break: lowest index, like jnp.argmax) -
#pragma unroll
    for (int off = 16; off > 0; off >>= 1) {
      float ov = __shfl_xor(bv, off, 32);
      int   oi = __shfl_xor(bi, off, 32);
      if (ov > bv || (ov == bv && oi < bi)) { bv = ov; bi = oi; }
    }
    const int lane = tid & 31, wv = tid >> 5;
    if (lane == 0) { sRv[wv] = bv; sRi[wv] = bi; }
    __syncthreads();

    // ---- wave 0: cross-wave combine (shuffle) + cross-WG split-barrier ----
    if (tid < 32) {
      float v = sRv[tid];
      int   idx = sRi[tid];
#pragma unroll
      for (int off = 16; off > 0; off >>= 1) {
        float ov = __shfl_xor(v, off, 32);
        int   oi = __shfl_xor(idx, off, 32);
        if (ov > v || (ov == v && oi < idx)) { v = ov; idx = oi; }
      }
      // lane 0 now holds the block winner.
      if (tid == 0) {
        int l = idx - base;                // winner is one of our own points
        float wx = sRaw[l * 3 + 0], wy = sRaw[l * 3 + 1], wz = sRaw[l * 3 + 2];
        // key: non-negative float bits are order-preserving; ~idx in low bits
        // makes the max key pick the smallest index on value ties.
        u64* sl = slots + (((size_t)(j & 1) * B_ + b) * SPLIT + split) * 4;
        u64 key = ((u64)__float_as_uint(v) << 32) | (u64)(0xFFFFFFFFu - (u32)idx);
        __hip_atomic_store(&sl[0], key,             __ATOMIC_RELAXED, __HIP_MEMORY_SCOPE_AGENT);
        __hip_atomic_store(&sl[1], pack2f(wx, wy),  __ATOMIC_RELAXED, __HIP_MEMORY_SCOPE_AGENT);
        __hip_atomic_store(&sl[2], pack2f(wz, 0.f), __ATOMIC_RELAXED, __HIP_MEMORY_SCOPE_AGENT);
        __hip_atomic_fetch_add(&cnt[b], 1u, __ATOMIC_RELEASE, __HIP_MEMORY_SCOPE_AGENT);
      }
      // lanes 0..7: arrive-spin (one vector load per poll for all 8 lanes),
      // then lane-parallel slot read + 8-lane shuffle max-key combine.
      if (tid < SPLIT) {
        const u32 target = (u32)(j + 1) * SPLIT;
        while (__hip_atomic_load(&cnt[b], __ATOMIC_ACQUIRE, __HIP_MEMORY_SCOPE_AGENT) < target)
          __builtin_amdgcn_s_sleep(1);
        u64* sb = slots + ((size_t)(j & 1) * B_ + b) * SPLIT * 4;
        u64 k2 = __hip_atomic_load(&sb[tid * 4 + 0], __ATOMIC_RELAXED, __HIP_MEMORY_SCOPE_AGENT);
        u64 xy = __hip_atomic_load(&sb[tid * 4 + 1], __ATOMIC_RELAXED, __HIP_MEMORY_SCOPE_AGENT);
        u64 zz = __hip_atomic_load(&sb[tid * 4 + 2], __ATOMIC_RELAXED, __HIP_MEMORY_SCOPE_AGENT);
        float nx = __uint_as_float((u32)xy);
        float ny = __uint_as_float((u32)(xy >> 32));
        float nz = __uint_as_float((u32)zz);
#pragma unroll
        for (int off = 4; off > 0; off >>= 1) {      // lanes 0..7 stay in-set
          u64   ok = __shfl_xor(k2, off, 32);
          float ox = __shfl_xor(nx, off, 32);
          float oy = __shfl_xor(ny, off, 32);
          float oz = __shfl_xor(nz, off, 32);
          if (ok > k2) { k2 = ok; nx = ox; ny = oy; nz = oz; }
        }
        if (tid == 0) { sC[0] = nx; sC[1] = ny; sC[2] = nz; }
      }
    }
    __syncthreads();
  }
}

extern "C" void kernel_launch(void* const* d_in, const int* in_sizes, int n_in,
                              void* d_out, int out_size, void* d_ws, size_t ws_size,
                              hipStream_t stream) {
  (void)in_sizes; (void)n_in; (void)out_size; (void)ws_size;
  const float* x = (const float*)d_in[0];
  float* out = (float*)d_out;

  // ws layout: [0,128)   : 32 u32 arrive counters (zeroed every call)
  //            [256, ...): 2(parity) x 32(batch) x 8(split) x 4 u64 slots (16KB)
  u32* cnt   = (u32*)d_ws;
  u64* slots = (u64*)((char*)d_ws + 256);

  // Opt in to >64KB dynamic LDS (gfx1250 WGP supports 320KB/workgroup).
  (void)hipFuncSetAttribute((const void*)fps_kernel,
                            hipFuncAttributeMaxDynamicSharedMemorySize,
                            LDS_BYTES);

  fps_init<<<1, 64, 0, stream>>>(cnt);
  dim3 grid(B_ * SPLIT);
  fps_kernel<<<grid, TPB, LDS_BYTES, stream>>>(x, out, cnt, slots);
}